// RelationalDelayGNNLiteStage_75041668595838
// MI455X (gfx1250) — compile-verified
//
#include <hip/hip_runtime.h>

// ---------------- problem constants (fixed by reference) ----------------
#define N_NODES 50000
#define N_EDGES 800000
#define DIM     64
#define T_LAYERS 4
#define E_TYPES  3
#define MAX_HOP  6   // T_LAYERS + 2

typedef __attribute__((ext_vector_type(2))) float v2f;
typedef __attribute__((ext_vector_type(8))) float v8f;

// ---------------- zero the 4 aggregation planes ----------------
__global__ __launch_bounds__(256) void zero_f4_kernel(float4* __restrict__ p, size_t n4) {
  size_t i = (size_t)blockIdx.x * blockDim.x + threadIdx.x;
  if (i < n4) p[i] = make_float4(0.f, 0.f, 0.f, 0.f);
}

// ---------------- single-pass edge scatter for layer t ----------------
// One wave (32 lanes, float2/lane = 64 channels) per edge; 8 edges per 256-thread block.
// agg layout: plane 0..2 = edge-type buckets, plane 3 = nu_kt-weighted hop bucket.
__global__ __launch_bounds__(256) void scatter_kernel(
    const int* __restrict__ esrc, const int* __restrict__ edst,
    const int* __restrict__ ehop, const int* __restrict__ etyp,
    const float* __restrict__ x_t,   // xs[t]
    const float* __restrict__ h0,    // xs[0]
    const float* __restrict__ h1,    // xs[1]
    const float* __restrict__ h2,    // xs[2]
    const float* __restrict__ nu_kt, // [T_LAYERS, MAX_HOP]
    int t,
    float* __restrict__ agg)         // [4, N_NODES, DIM]
{
  const int lane = threadIdx.x & 31;
  const int e = blockIdx.x * (blockDim.x >> 5) + (threadIdx.x >> 5);
  if (e >= N_EDGES) return;

  const int s  = esrc[e];
  const int d  = edst[e];
  const int ty = etyp[e];
  const int hp = ehop[e];

  // edge-type relational pass: agg[ty][d] += x_t[s]
  const v2f vx = ((const v2f*)(x_t + (size_t)s * DIM))[lane];
  float* base = agg + ((size_t)ty * N_NODES + (size_t)d) * DIM + lane * 2;
  unsafeAtomicAdd(base + 0, vx.x);
  unsafeAtomicAdd(base + 1, vx.y);

  // delayed multi-hop pass (k in [2, t+1]): agg[3][d] += nu_kt[t,hp] * xs[t-hp+1][s]
  if (hp >= 2 && hp <= t + 1) {
    const int idx = t - hp + 1;                       // 0..t-1
    const float* xh = (idx == 0) ? h0 : (idx == 1) ? h1 : h2;
    const float scale = nu_kt[t * MAX_HOP + hp];
    const v2f vh = ((const v2f*)(xh + (size_t)s * DIM))[lane];
    float* hbase = agg + ((size_t)3 * N_NODES + (size_t)d) * DIM + lane * 2;
    unsafeAtomicAdd(hbase + 0, scale * vh.x);
    unsafeAtomicAdd(hbase + 1, scale * vh.y);
  }
}

// ---------------- fused 4x GEMM + bias + ReLU + residual (WMMA fp32) ----------------
// Block = 128 threads = 4 waves. Wave w computes the 16x16 tile
// rows [16*blockIdx.x, +16) x cols [16*w, +16).  50000 % 16 == 0 -> no guards.
// x_out = x_in + relu( sum_s nu_s * (agg_s @ W_s) + bias_t )
__global__ __launch_bounds__(128) void gemm_relu_kernel(
    const float* __restrict__ agg,     // [4, N_NODES, DIM]
    const float* __restrict__ W_edge,  // [E_TYPES, T_LAYERS, DIM, DIM]
    const float* __restrict__ b_edge,  // [E_TYPES, T_LAYERS, DIM]
    const float* __restrict__ nu_edge, // [E_TYPES, T_LAYERS]
    const float* __restrict__ W_t,     // [T_LAYERS, DIM, DIM]
    const float* __restrict__ b_t,     // [T_LAYERS, DIM]
    const float* __restrict__ nu_kt,   // [T_LAYERS, MAX_HOP]
    const float* __restrict__ x_in,    // xs[t]
    float* __restrict__ x_out,         // xs[t+1] (or d_out)
    int t)
{
  const int lane    = threadIdx.x & 31;
  const int wave    = threadIdx.x >> 5;
  const int rowBase = blockIdx.x * 16;
  const int colBase = wave * 16;
  const int n  = lane & 15;   // A-frag M index and B/C-frag N index (ISA layout)
  const int kh = lane >> 4;   // K-half selector for A/B frags; C-frag M-half

  v8f c = {};                 // nu-weighted accumulation over the 4 sources
  #pragma unroll
  for (int s = 0; s < 4; ++s) {
    const float* A = agg + (size_t)s * N_NODES * DIM + (size_t)(rowBase + n) * DIM;
    const float* W = (s < 3) ? (W_edge + ((size_t)s * T_LAYERS + t) * DIM * DIM)
                             : (W_t   + (size_t)t * DIM * DIM);
    v8f cs = {};
    #pragma unroll
    for (int k0 = 0; k0 < DIM; k0 += 4) {   // 16 chained K=4 WMMA steps -> K=64
      const int ka = k0 + 2 * kh;
      v2f a, b;
      a.x = A[ka];
      a.y = A[ka + 1];
      b.x = W[(size_t)ka       * DIM + colBase + n];
      b.y = W[(size_t)(ka + 1) * DIM + colBase + n];
      cs = __builtin_amdgcn_wmma_f32_16x16x4_f32(
          /*neg_a=*/false, a, /*neg_b=*/false, b,
          /*c_mod=*/(short)0, cs, /*reuse_a=*/false, /*reuse_b=*/false);
    }
    const float nu = (s < 3) ? nu_edge[s * T_LAYERS + t] : 1.0f;
    c += nu * cs;
  }

  // combined bias: sum_e nu_edge[e,t]*b_edge[e,t] + (sum_{k=2..t+1} nu_kt[t,k])*b_t[t]
  const int col = colBase + n;
  float bias = 0.0f;
  #pragma unroll
  for (int e2 = 0; e2 < E_TYPES; ++e2)
    bias += nu_edge[e2 * T_LAYERS + t] * b_edge[(e2 * T_LAYERS + t) * DIM + col];
  for (int k = 2; k <= t + 1; ++k)
    bias += nu_kt[t * MAX_HOP + k] * b_t[t * DIM + col];

  #pragma unroll
  for (int r = 0; r < 8; ++r) {
    const int row = rowBase + r + 8 * kh;   // C/D layout: VGPR r holds M=r (lanes 0-15), M=r+8 (16-31)
    float v = c[r] + bias;
    v = v > 0.0f ? v : 0.0f;                // ReLU
    x_out[(size_t)row * DIM + col] = x_in[(size_t)row * DIM + col] + v;  // residual
  }
}

// ---------------- host orchestration ----------------
extern "C" void kernel_launch(void* const* d_in, const int* in_sizes, int n_in,
                              void* d_out, int out_size, void* d_ws, size_t ws_size,
                              hipStream_t stream) {
  const float* x       = (const float*)d_in[0];
  const float* W_edge  = (const float*)d_in[1];
  const float* b_edge  = (const float*)d_in[2];
  const float* nu_edge = (const float*)d_in[3];
  const float* W_t     = (const float*)d_in[4];
  const float* b_t     = (const float*)d_in[5];
  const float* nu_kt   = (const float*)d_in[6];
  const int*   esrc    = (const int*)d_in[7];
  const int*   edst    = (const int*)d_in[8];
  const int*   ehop    = (const int*)d_in[9];
  const int*   etyp    = (const int*)d_in[10];

  // workspace: agg[4][N][64] then xs1, xs2, xs3 history planes (~90 MB total)
  float* agg = (float*)d_ws;
  float* xs1 = agg + (size_t)4 * N_NODES * DIM;
  float* xs2 = xs1 + (size_t)N_NODES * DIM;
  float* xs3 = xs2 + (size_t)N_NODES * DIM;

  float* xptr[5] = { const_cast<float*>(x), xs1, xs2, xs3, (float*)d_out };

  const size_t n4 = (size_t)4 * N_NODES * DIM / 4;
  const int zeroBlocks    = (int)((n4 + 255) / 256);
  const int scatterBlocks = (N_EDGES + 7) / 8;   // 8 edges (waves) per block
  const int gemmBlocks    = N_NODES / 16;        // 3125, exact

  for (int t = 0; t < T_LAYERS; ++t) {
    zero_f4_kernel<<<zeroBlocks, 256, 0, stream>>>((float4*)agg, n4);
    scatter_kernel<<<scatterBlocks, 256, 0, stream>>>(
        esrc, edst, ehop, etyp,
        xptr[t], xptr[0], xptr[1], xptr[2],
        nu_kt, t, agg);
    gemm_relu_kernel<<<gemmBlocks, 128, 0, stream>>>(
        agg, W_edge, b_edge, nu_edge, W_t, b_t, nu_kt,
        xptr[t], xptr[t + 1], t);
  }
}